// OxideModel_30683246363061
// MI455X (gfx1250) — compile-verified
//
#include <hip/hip_runtime.h>
#include <math.h>

namespace {

constexpr float kGamma = 0.5772156649015329f;

// Ei(z) for z < 0, branchless (matches reference jnp.where semantics).
__device__ __forceinline__ float ei_neg(float z) {
  float x = -z;  // x > 0
  bool small_x = x < 1.0f;

  // --- series branch: E1(x) = -gamma - ln(x) - sum_k (-x)^k / (k*k!) ---
  float xs = small_x ? x : 0.5f;   // safe value for unused lanes
  float nxs = -xs;
  float term = 1.0f;
  float acc = 0.0f;
#pragma unroll
  for (int k = 1; k <= 25; ++k) {
    const float rk = 1.0f / (float)k;   // compile-time constant
    term = term * nxs * rk;             // term = (-x)^k / k!
    acc = fmaf(term, -rk, acc);         // acc -= term / k
  }
  float e1_small = -kGamma - logf(xs) + acc;

  // --- rational branch (A&S 5.1.56): E1(x) = exp(-x)/x * P4(x)/Q4(x) ---
  float xl = small_x ? 2.0f : x;
  float t1 = xl + 8.5733287401f;
  float t2 = fmaf(t1, xl, 18.0590169730f);
  float t3 = fmaf(t2, xl, 8.6347608925f);
  float num = fmaf(t3, xl, 0.2677737343f);
  float u1 = xl + 9.5733223454f;
  float u2 = fmaf(u1, xl, 25.6329561486f);
  float u3 = fmaf(u2, xl, 21.0996530827f);
  float den = fmaf(u3, xl, 3.9584969228f);
  float e1_large = expf(-xl) / xl * (num / den);

  float e1 = small_x ? e1_small : e1_large;
  return -e1;
}

struct Params {
  float E, U, K, fT, expK, intTmax, c1;
};

__device__ __forceinline__ float oxide_elem(float t, const Params& p) {
  float Eot = p.E / t;                                   // E / t
  float ei = ei_neg(-Eot);                               // Ei(-E/t)
  float integ = p.expK * fmaf(t, expf(-Eot), p.E * ei);  // integral(t)
  float result = integ - p.intTmax;                      // integral(t)-integral(Tmax)
  float r = fmaxf(0.0f, fmaf(-p.c1, result, p.U));       // relu(U - c1*result)
  float e = expf((p.K - Eot) - p.fT);                    // exp(f(t) - f(Tmax))
  return e * r * r;
}

__global__ void __launch_bounds__(256)
oxide_model_kernel(const float* __restrict__ in,
                   const float* __restrict__ gshift,
                   const float* __restrict__ e_param,
                   const float* __restrict__ tmax_delta,
                   const float* __restrict__ v_max,
                   float* __restrict__ out, int n) {
  // ---- uniform scalar prologue (hoisted; scalar loads via s_load) ----
  Params p;
  float E = expf(e_param[0]);                         // E_STATE scale == 1
  float V = expf(v_max[0]);                           // V_STATE scale == 1
  float sig = 1.0f / (1.0f + expf(-tmax_delta[0]));   // delta_scale == 1
  float T = 1000.0f + 100.0f * (2.0f * sig - 1.0f) + gshift[0];
  float U = sqrtf(V);
  float K = E / T + (2.0f / 3.0f) * logf(1.5f * E * U / (T * T));
  p.E = E;
  p.U = U;
  p.K = K;
  p.fT = K - E / T;
  p.expK = expf(K);
  p.intTmax = p.expK * fmaf(T, expf(-E / T), E * ei_neg(-E / T));
  p.c1 = (1.0f / 3.0f) * expf(0.5f * p.fT);

  // ---- streaming body: b128 loads/stores + speculative prefetch ----
  int n4 = n >> 2;
  const float4* __restrict__ in4 = (const float4*)in;
  float4* __restrict__ out4 = (float4*)out;
  int stride = (int)(gridDim.x * blockDim.x);
  for (int i = (int)(blockIdx.x * blockDim.x + threadIdx.x); i < n4; i += stride) {
    // Prefetch next grid-stride iteration's line (global_prefetch_b8).
    // Speculative: OOB addresses are silently dropped by hardware.
    __builtin_prefetch(in4 + i + stride, 0, 0);
    float4 t = in4[i];
    float4 o;
    o.x = oxide_elem(t.x, p);
    o.y = oxide_elem(t.y, p);
    o.z = oxide_elem(t.z, p);
    o.w = oxide_elem(t.w, p);
    out4[i] = o;
  }

  // ---- tail (n % 4), scalar ----
  int base = n4 << 2;
  for (int i = base + (int)(blockIdx.x * blockDim.x + threadIdx.x); i < n;
       i += stride) {
    out[i] = oxide_elem(in[i], p);
  }
}

}  // namespace

extern "C" void kernel_launch(void* const* d_in, const int* in_sizes, int n_in,
                              void* d_out, int out_size, void* d_ws, size_t ws_size,
                              hipStream_t stream) {
  const float* in = (const float*)d_in[0];         // input [N] fp32
  const float* gshift = (const float*)d_in[1];     // global_shift [1]
  const float* e_param = (const float*)d_in[2];    // E_param [1]
  const float* tmax_delta = (const float*)d_in[3]; // T_max_delta [1]
  const float* v_max = (const float*)d_in[4];      // V_max [1]
  float* out = (float*)d_out;

  int n = in_sizes[0];
  int n4 = n >> 2;

  const int block = 256;  // 8 wave32 waves per workgroup
  // ~4 float4 iterations per thread: 4096 blocks for N=16.7M -> 32K waves.
  long long threads_wanted = ((long long)n4 + 3) / 4;
  int grid = (int)((threads_wanted + block - 1) / block);
  if (grid < 1) grid = 1;

  oxide_model_kernel<<<grid, block, 0, stream>>>(in, gshift, e_param,
                                                 tmax_delta, v_max, out, n);
}